// MemoryBank_52432960749632
// MI455X (gfx1250) — compile-verified
//
#include <hip/hip_runtime.h>
#include <cstdint>

// ---------------------------------------------------------------------------
// MemoryBank op for MI455X (gfx1250).
// Pure data movement: ~800 MB of stores, ~25 MFLOP -> store-bandwidth bound.
// Broadcast uses the CDNA5 async global<->LDS DMA path (ASYNCcnt).
// NOTE: mb_broadcast is defined FIRST so the disasm snippet shows the
//       global_load_async_to_lds_b128 / global_store_async_from_lds_b128 ops.
// ---------------------------------------------------------------------------

typedef float f4 __attribute__((ext_vector_type(4)));
typedef int   v4i __attribute__((vector_size(16)));   // matches builtin param

#define B_    64
#define C_    1501
#define D_    2048
#define ROW4  512            // float4 per row
#define TPB   256            // 8 waves (wave32)
#define ALPHA_ 0.01f
#define ONEMA_ 0.99f
#define EPS_   1e-7f

#if defined(__HIP_DEVICE_COMPILE__)
# if __has_builtin(__builtin_amdgcn_global_load_async_to_lds_b128) && \
     __has_builtin(__builtin_amdgcn_global_store_async_from_lds_b128) && \
     __has_builtin(__builtin_amdgcn_s_wait_asynccnt)
#  define ASYNC_OK 1
# else
#  define ASYNC_OK 0
# endif
#else
# define ASYNC_OK 0
#endif

#if ASYNC_OK
typedef __attribute__((address_space(1))) v4i* gas_p;   // global (AS1)
typedef __attribute__((address_space(3))) v4i* las_p;   // LDS    (AS3)

__device__ __forceinline__ void async_g2l_b128(void* lds, const void* g) {
  // global -> LDS, 16B per lane, tracked on ASYNCcnt
  __builtin_amdgcn_global_load_async_to_lds_b128(
      (gas_p)(uintptr_t)g, (las_p)(uint32_t)(uintptr_t)lds, 0, 0);
}
__device__ __forceinline__ void async_l2g_b128(void* g, const void* lds) {
  // LDS -> global, 16B per lane, tracked on ASYNCcnt (bypasses VGPRs)
  __builtin_amdgcn_global_store_async_from_lds_b128(
      (gas_p)(uintptr_t)g, (las_p)(uint32_t)(uintptr_t)lds, 0, 0);
}
#endif

// --- tmp_repeated: 64 copies of (on-the-fly) tmp_memory --------------------
// One block per 2 rows (exactly 16 KB). Async path: stage rows in LDS once,
// then 64 LDS->global async stores; data never round-trips through VGPRs.
__global__ void __launch_bounds__(TPB) mb_broadcast(
    const float* __restrict__ x, const float* __restrict__ mem,
    const int* __restrict__ owner, float* __restrict__ out_rep) {
  const int t = threadIdx.x;
  const int c0 = blockIdx.x * 2;
  const int nrows = (c0 + 1 < C_) ? 2 : 1;    // C_ is odd -> last block: 1 row
  f4* rep = (f4*)out_rep;
  const long base = (long)c0 * ROW4;
  const long cd4  = (long)C_ * ROW4;

#if ASYNC_OK
  __shared__ f4 lbuf[2 * ROW4];               // 16 KB
  for (int r = 0; r < nrows; ++r) {
    const int c = c0 + r;
    const int b = owner[c];
    const f4* src = (b >= 0) ? (const f4*)x + (size_t)b * ROW4
                             : (const f4*)mem + (size_t)c * ROW4;
    async_g2l_b128(&lbuf[r * ROW4 + t],       src + t);
    async_g2l_b128(&lbuf[r * ROW4 + TPB + t], src + TPB + t);
  }
  __builtin_amdgcn_s_wait_asynccnt(0);
  __syncthreads();

  const int nf4 = nrows * ROW4;
  for (int bb = 0; bb < B_; ++bb) {
    f4* dst = rep + (long)bb * cd4 + base;
    for (int i = t; i < nf4; i += TPB)
      async_l2g_b128(dst + i, &lbuf[i]);
  }
  __builtin_amdgcn_s_wait_asynccnt(0);
#else
  f4 v[4];
  for (int r = 0; r < nrows; ++r) {
    const int c = c0 + r;
    const int b = owner[c];
    const f4* src = (b >= 0) ? (const f4*)x + (size_t)b * ROW4
                             : (const f4*)mem + (size_t)c * ROW4;
    v[2 * r]     = src[t];
    v[2 * r + 1] = src[TPB + t];
  }
  for (int bb = 0; bb < B_; ++bb) {
    f4* dst = rep + (long)bb * cd4 + base;
    for (int r = 0; r < nrows; ++r) {
      // NT hint: 787 MB stream, keep it out of L2 (L2 holds the 12 MB source)
      __builtin_nontemporal_store(v[2 * r],     dst + r * ROW4 + t);
      __builtin_nontemporal_store(v[2 * r + 1], dst + r * ROW4 + TPB + t);
    }
  }
#endif
}

// --- owner table: owner[c] = max b with cls[b]==c, else -1 -----------------
__global__ void mb_owner_init(int* __restrict__ owner) {
  int c = blockIdx.x * blockDim.x + threadIdx.x;
  if (c < C_) owner[c] = -1;
}

__global__ void mb_owner_scatter(const int* __restrict__ cls,
                                 int* __restrict__ owner) {
  int b = threadIdx.x;
  if (b < B_) atomicMax(&owner[cls[b]], b);   // last-write-wins == max b
}

// --- selected = memory[cls_list]  (gather of ORIGINAL memory) --------------
__global__ void __launch_bounds__(TPB) mb_selected(
    const float* __restrict__ mem, const int* __restrict__ cls,
    float* __restrict__ sel) {
  int i = blockIdx.x * TPB + threadIdx.x;     // 0 .. B_*ROW4-1
  int b = i >> 9, d4 = i & (ROW4 - 1);
  ((f4*)sel)[i] = ((const f4*)mem)[(size_t)cls[b] * ROW4 + d4];
}

// --- new_mem: row-wise EMA + L2 normalize (one block per row) --------------
__global__ void __launch_bounds__(TPB) mb_newmem(
    const float* __restrict__ x, const float* __restrict__ mem,
    const int* __restrict__ owner, float* __restrict__ out_new) {
  const int c = blockIdx.x;
  const int t = threadIdx.x;
  const int b = owner[c];
  const f4* mrow = (const f4*)mem + (size_t)c * ROW4;
  const f4* trow = (b >= 0) ? (const f4*)x + (size_t)b * ROW4 : mrow;

  f4 v0 = mrow[t]       * ONEMA_ + trow[t]       * ALPHA_;
  f4 v1 = mrow[t + TPB] * ONEMA_ + trow[t + TPB] * ALPHA_;

  float ss = v0.x*v0.x + v0.y*v0.y + v0.z*v0.z + v0.w*v0.w
           + v1.x*v1.x + v1.y*v1.y + v1.z*v1.z + v1.w*v1.w;

  __shared__ float red[TPB];
  red[t] = ss;
  __syncthreads();
  for (int s = TPB / 2; s > 0; s >>= 1) {
    if (t < s) red[t] += red[t + s];
    __syncthreads();
  }
  const float inv = 1.0f / (sqrtf(red[0]) + EPS_);

  f4* orow = (f4*)out_new + (size_t)c * ROW4;
  orow[t]       = v0 * inv;
  orow[t + TPB] = v1 * inv;
}

// ---------------------------------------------------------------------------
extern "C" void kernel_launch(void* const* d_in, const int* in_sizes, int n_in,
                              void* d_out, int out_size, void* d_ws, size_t ws_size,
                              hipStream_t stream) {
  (void)in_sizes; (void)n_in; (void)out_size; (void)ws_size;

  const float* x   = (const float*)d_in[0];   // [B, D]
  const int*   cls = (const int*)d_in[1];     // [B]
  const float* mem = (const float*)d_in[2];   // [C, D]

  float* out     = (float*)d_out;
  float* out_sel = out;                                       // [B, D]
  float* out_rep = out + (size_t)B_ * D_;                     // [B, C, D]
  float* out_new = out_rep + (size_t)B_ * C_ * D_;            // [C, D]

  int* owner = (int*)d_ws;                                    // C_ ints

  mb_owner_init   <<<(C_ + TPB - 1) / TPB, TPB, 0, stream>>>(owner);
  mb_owner_scatter<<<1, B_, 0, stream>>>(cls, owner);
  mb_selected     <<<(B_ * ROW4) / TPB, TPB, 0, stream>>>(mem, cls, out_sel);
  mb_newmem       <<<C_, TPB, 0, stream>>>(x, mem, owner, out_new);
  mb_broadcast    <<<(C_ + 1) / 2, TPB, 0, stream>>>(x, mem, owner, out_rep);
}